// ModelNew_58128087384663
// MI455X (gfx1250) — compile-verified
//
#include <hip/hip_runtime.h>

typedef __attribute__((ext_vector_type(2))) float v2f;
typedef __attribute__((ext_vector_type(8))) float v8f;

#define MM 65536
#define KK 4096
#define NN 4096

// ---------------- kernel 0: zero the workspace accumulators ----------------
__global__ void ModelNew_zero_kernel(float* __restrict__ p, int n) {
    int i = blockIdx.x * blockDim.x + threadIdx.x;
    if (i < n) p[i] = 0.0f;
}

// ---------------- kernel 1: w_sum[k] = sum_n W[n,k] ----------------
// grid = (KK/256, 8), block = 256.  Each block sums 512 rows of a 256-col slab.
// Fully coalesced (256 adjacent lanes read 1 KB contiguous per step).
__global__ void ModelNew_colsum_kernel(const float* __restrict__ w,
                                       float* __restrict__ wsum) {
    const int col = blockIdx.x * blockDim.x + threadIdx.x;
    const int n0  = blockIdx.y * (NN / 8);
    const float* p = w + (size_t)n0 * KK + col;
    float s = 0.0f;
#pragma unroll 8
    for (int n = 0; n < NN / 8; ++n)
        s += p[(size_t)n * KK];
    atomicAdd(&wsum[col], s);
}

// ---------------- kernel 2: b_sum = sum(bias) ----------------
__global__ void ModelNew_biassum_kernel(const float* __restrict__ b,
                                        float* __restrict__ bsum) {
    float s = 0.0f;
    for (int i = threadIdx.x; i < NN; i += blockDim.x) s += b[i];
    atomicAdd(bsum, s);
}

// ---------------- kernel 3: y = x @ w_sum + b_sum via V_WMMA_F32_16X16X4_F32 -
// 256 threads = 8 waves; each wave owns a 16-row tile (128 rows / block).
// A (16x4 f32): lanes 0-15 hold M=lane, K={0,1}; lanes 16-31 hold M=lane-16,
// K={2,3}  ->  per-lane global_load_b64 of x[row, k+kOff .. +1].
// B: broadcast the same w_sum pair to every lane -> every C column equals the
// desired dot product; no masking needed in the hot loop.
// C (16x16 f32): lane 0 VGPR i = (M=i, N=0); lane 16 VGPR i = (M=8+i, N=0).
__global__ void __launch_bounds__(256, 1)
ModelNew_gemv_wmma_kernel(const float* __restrict__ x,
                          const float* __restrict__ wsum,
                          const float* __restrict__ bsum,
                          float* __restrict__ out) {
    const int lane = threadIdx.x & 31;
    const int wave = threadIdx.x >> 5;
    const int r0   = blockIdx.x * 128 + wave * 16;
    const int row  = r0 + (lane & 15);
    const int kOff = (lane >> 4) << 1;           // 0 for lanes 0-15, 2 for 16-31

    const float* xp = x + (size_t)row * KK + kOff;
    const float* wp = wsum + kOff;

    v8f c = {0.f, 0.f, 0.f, 0.f, 0.f, 0.f, 0.f, 0.f};

#pragma unroll 8
    for (int k = 0; k < KK; k += 4) {
        v2f a = *(const v2f*)(xp + k);           // 8B-aligned global_load_b64
        v2f b = *(const v2f*)(wp + k);           // hits L0/L2 (16 KB total)
        // 8 args: (neg_a, A, neg_b, B, c_mod, C, reuse_a, reuse_b)
        c = __builtin_amdgcn_wmma_f32_16x16x4_f32(false, a, false, b,
                                                  (short)0, c, false, false);
    }

    const float bb = *bsum;
    if ((lane & 15) == 0) {
        const int base = r0 + ((lane >> 4) << 3);  // lane0 -> rows 0-7, lane16 -> 8-15
#pragma unroll
        for (int i = 0; i < 8; ++i)
            out[base + i] = c[i] + bb;
    }
}

extern "C" void kernel_launch(void* const* d_in, const int* in_sizes, int n_in,
                              void* d_out, int out_size, void* d_ws, size_t ws_size,
                              hipStream_t stream) {
    const float* x    = (const float*)d_in[0];   // (M, K) f32
    const float* w    = (const float*)d_in[1];   // (N, K) f32
    const float* bias = (const float*)d_in[2];   // (N,)  f32
    float* out  = (float*)d_out;                 // (M, 1) f32
    float* wsum = (float*)d_ws;                  // K floats
    float* bsum = wsum + KK;                     // 1 float

    // workspace is poisoned -> zero accumulators every call (deterministic)
    ModelNew_zero_kernel<<<(KK + 1 + 255) / 256, 256, 0, stream>>>(wsum, KK + 1);
    ModelNew_colsum_kernel<<<dim3(KK / 256, 8), 256, 0, stream>>>(w, wsum);
    ModelNew_biassum_kernel<<<1, 256, 0, stream>>>(bias, bsum);
    ModelNew_gemv_wmma_kernel<<<MM / 128, 256, 0, stream>>>(x, wsum, bsum, out);
}